// GraphNorm_58866821759677
// MI455X (gfx1250) — compile-verified
//
#include <hip/hip_runtime.h>

// GraphNorm for MI455X (gfx1250, wave32).
// Memory-bound: ~1 GB effective HBM traffic with per-segment fusion
// (segment ~125 KB; phase-2 re-read of x hits L2).
// Phase-1 stats use V_WMMA_F32_16X16X4_F32 with an all-ones A matrix
// (colsum = 1^T X reduction); phase-2 streams float4 with NT hints.

typedef __attribute__((ext_vector_type(2))) float v2f;
typedef __attribute__((ext_vector_type(4))) float v4f;
typedef __attribute__((ext_vector_type(8))) float v8f;

#define NUM_GRAPHS 4096
#define HIDDEN 128
#define GN_EPS 1e-5f

// ---------------------------------------------------------------------------
// Kernel 1: segment boundaries via binary search on the sorted batch array.
// starts[g] = first index i with batch[i] >= g ; starts[NUM_GRAPHS] = N.
// ---------------------------------------------------------------------------
__global__ void gn_find_starts(const int* __restrict__ batch, int n,
                               int* __restrict__ starts) {
    int g = blockIdx.x * blockDim.x + threadIdx.x;
    if (g > NUM_GRAPHS) return;
    int lo = 0, hi = n;
    while (lo < hi) {
        int mid = (lo + hi) >> 1;
        if (batch[mid] < g) lo = mid + 1; else hi = mid;
    }
    starts[g] = lo;
}

// ---------------------------------------------------------------------------
// Kernel 2: one 256-thread block (8 wave32) per graph segment.
//   Phase 1: WMMA ones-reduction -> per-column sum / sumsq.
//            Main loop = full 8-row tiles, no per-lane branching;
//            tails peeled (one 4-row tile, then one masked tile).
//   Phase 2: nontemporal float4 streaming normalize.
// ---------------------------------------------------------------------------
__launch_bounds__(256)
__global__ void gn_main(const float* __restrict__ x,
                        const int* __restrict__ starts,
                        const float* __restrict__ weight,
                        const float* __restrict__ bias,
                        float* __restrict__ out) {
    __shared__ float s_sum[HIDDEN];
    __shared__ float s_sq[HIDDEN];
    __shared__ v4f   s_scale4[HIDDEN / 4];
    __shared__ v4f   s_shift4[HIDDEN / 4];
    float* s_scale = (float*)s_scale4;
    float* s_shift = (float*)s_shift4;

    const int g    = blockIdx.x;
    const int row0 = starts[g];
    const int row1 = starts[g + 1];
    const int cnt  = row1 - row0;

    const int wave = threadIdx.x >> 5;   // 0..7, each owns 16 columns
    const int lane = threadIdx.x & 31;
    const int n16  = lane & 15;          // column within the wave's stripe
    const int krow = lane >> 4;          // 0 or 1: which K-row this lane feeds
    const int colBase = wave * 16;

    // ---- Phase 1: sum / sumsq over the segment via V_WMMA_F32_16X16X4_F32.
    // A = all-ones 16x4  =>  D[m][n] = sum_k B[k][n]  (every row = colsum).
    // B (4x16) lane map: VGPR0 = {K0 lanes 0-15 | K1 lanes 16-31},
    //                    VGPR1 = {K2 lanes 0-15 | K3 lanes 16-31}.
    v2f a; a.x = 1.0f; a.y = 1.0f;
    v8f accS0 = {}, accS1 = {};          // column sums (two chains)
    v8f accQ0 = {}, accQ1 = {};          // column sums of squares

    const float* __restrict__ xc = x + colBase + n16;
    const int end8 = row0 + (cnt & ~7);  // last full 8-row super-tile
    const int end4 = row0 + (cnt & ~3);  // last full 4-row tile

    // Main loop: 8 rows / iteration, 4 independent loads in flight,
    // two accumulator chains so the 4 WMMAs have no C dependency between
    // tile0 and tile1.
    const float* p = xc + (size_t)(row0 + krow) * HIDDEN;
    int r = row0;
    for (; r < end8; r += 8, p += 8 * HIDDEN) {
        const float b0 = p[0 * HIDDEN];
        const float b1 = p[2 * HIDDEN];
        const float c0 = p[4 * HIDDEN];
        const float c1 = p[6 * HIDDEN];
        v2f b;  b.x  = b0;      b.y  = b1;
        v2f bq; bq.x = b0 * b0; bq.y = b1 * b1;
        v2f c;  c.x  = c0;      c.y  = c1;
        v2f cq; cq.x = c0 * c0; cq.y = c1 * c1;
        accS0 = __builtin_amdgcn_wmma_f32_16x16x4_f32(
            false, a, false, b,  (short)0, accS0, false, false);
        accQ0 = __builtin_amdgcn_wmma_f32_16x16x4_f32(
            false, a, false, bq, (short)0, accQ0, false, false);
        accS1 = __builtin_amdgcn_wmma_f32_16x16x4_f32(
            false, a, false, c,  (short)0, accS1, false, false);
        accQ1 = __builtin_amdgcn_wmma_f32_16x16x4_f32(
            false, a, false, cq, (short)0, accQ1, false, false);
    }
    // Mid tail: one full 4-row tile.
    if (r < end4) {
        const float b0 = p[0 * HIDDEN];
        const float b1 = p[2 * HIDDEN];
        v2f b;  b.x  = b0;      b.y  = b1;
        v2f bq; bq.x = b0 * b0; bq.y = b1 * b1;
        accS0 = __builtin_amdgcn_wmma_f32_16x16x4_f32(
            false, a, false, b,  (short)0, accS0, false, false);
        accQ0 = __builtin_amdgcn_wmma_f32_16x16x4_f32(
            false, a, false, bq, (short)0, accQ0, false, false);
        r += 4;
        p += 4 * HIDDEN;
    }
    // Final masked tail (<=3 rows): per-lane predicated loads, EXEC restored
    // before the WMMAs execute.
    if (r < row1) {
        const int ra = r + krow;
        const int rb = r + 2 + krow;
        const float b0 = (ra < row1) ? xc[(size_t)ra * HIDDEN] : 0.0f;
        const float b1 = (rb < row1) ? xc[(size_t)rb * HIDDEN] : 0.0f;
        v2f b;  b.x  = b0;      b.y  = b1;
        v2f bq; bq.x = b0 * b0; bq.y = b1 * b1;
        accS0 = __builtin_amdgcn_wmma_f32_16x16x4_f32(
            false, a, false, b,  (short)0, accS0, false, false);
        accQ0 = __builtin_amdgcn_wmma_f32_16x16x4_f32(
            false, a, false, bq, (short)0, accQ0, false, false);
    }

    const v8f accS = accS0 + accS1;
    const v8f accQ = accQ0 + accQ1;

    // D row M=0 lives in acc[0], lanes 0..15 (N = lane).
    if (lane < 16) {
        s_sum[colBase + n16] = accS[0];
        s_sq [colBase + n16] = accQ[0];
    }
    __syncthreads();

    // ---- Finalize per-column mean/var; fold weight/bias into scale/shift.
    if (threadIdx.x < HIDDEN) {
        const int d = threadIdx.x;
        const float c    = (float)cnt;
        const float safe = (c > 1.0f) ? c : 1.0f;
        float mean = s_sum[d] / safe;
        float var  = s_sq[d] / safe - mean * mean;
        if (!(c > 1.0f)) { mean = 0.0f; var = 1.0f; }  // torch init for <=1 node
        const float rstd = rsqrtf(var + GN_EPS);
        const float sc   = weight[d] * rstd;
        s_scale[d] = sc;
        s_shift[d] = bias[d] - mean * sc;
    }
    __syncthreads();

    // ---- Phase 2: streaming normalize. One wave per row, v4f per lane
    // (global_load_b128 / global_store_b128, 512 B per row per wave).
    // x lines are last-use here and out is never re-read -> NT hints keep
    // L2 capacity for other blocks' phase-1 segments.
    const v4f* __restrict__ x4 = reinterpret_cast<const v4f*>(x);
    v4f* __restrict__       o4 = reinterpret_cast<v4f*>(out);
    const v4f sc = s_scale4[lane];       // one ds_load_b128 each
    const v4f sh = s_shift4[lane];

    for (int rr = row0 + wave; rr < row1; rr += 8) {
        const size_t idx = (size_t)rr * (HIDDEN / 4) + lane;
        const v4f v = __builtin_nontemporal_load(&x4[idx]);
        v4f o;
        o.x = fmaf(v.x, sc.x, sh.x);
        o.y = fmaf(v.y, sc.y, sh.y);
        o.z = fmaf(v.z, sc.z, sh.z);
        o.w = fmaf(v.w, sc.w, sh.w);
        __builtin_nontemporal_store(o, &o4[idx]);
    }
}

// ---------------------------------------------------------------------------
// Host-side launcher (graph-capture safe: only kernel launches on `stream`).
// Inputs (setup_inputs order): x[f32 N*128], batch[int N], weight[f32 128],
// bias[f32 128]. d_ws holds the (NUM_GRAPHS+1) int starts array.
// ---------------------------------------------------------------------------
extern "C" void kernel_launch(void* const* d_in, const int* in_sizes, int n_in,
                              void* d_out, int out_size, void* d_ws, size_t ws_size,
                              hipStream_t stream) {
    const float* x      = (const float*)d_in[0];
    const int*   batch  = (const int*)d_in[1];
    const float* weight = (const float*)d_in[2];
    const float* bias   = (const float*)d_in[3];
    float*       out    = (float*)d_out;
    int*         starts = (int*)d_ws;   // NUM_GRAPHS + 1 ints

    const int n = in_sizes[1];          // number of nodes

    const int t1 = 256;
    const int b1 = (NUM_GRAPHS + 1 + t1 - 1) / t1;
    gn_find_starts<<<b1, t1, 0, stream>>>(batch, n, starts);

    gn_main<<<NUM_GRAPHS, 256, 0, stream>>>(x, starts, weight, bias, out);
}